// SSM_67267777790557
// MI455X (gfx1250) — compile-verified
//
#include <hip/hip_runtime.h>

// ---------------------------------------------------------------------------
// SSM scan on MI455X (gfx1250): chunked parallel scan with f32 WMMA + TDM.
//   drive = (HEAT*M*DT) @ Bw^T + D @ Ew^T            (parallel, kernel k_drive)
//   x_t   = x_{t-1} @ A^T + drive_t                  (3-phase chunked scan)
//   Y     = X @ Cw^T                                 (fused in emit phase)
// A (256KB) staged into LDS via tensor_load_to_lds; state (64KB) in LDS.
// ---------------------------------------------------------------------------

#define T_DIM 2048
#define B_DIM 64
#define NX 256
#define NU 64
#define ND 64
#define NY 64
#define NCHUNK 64
#define CHUNK_LEN 32   // NCHUNK * CHUNK_LEN == T_DIM

#define HEAT_CONST 1.1591509722222224f
#define XMIN_C 0.0f
#define XMAX_C 35.0f
#define UMIN_C (-5000.0f)
#define UMAX_C 5000.0f
#define DXMIN_C (-0.5f)
#define DXMAX_C 0.5f
#define Q_DX_C 100.0f
#define Q_DX_UD_C 100000.0f
#define Q_CON_X_C 10.0f
#define Q_CON_U_C 10.0f

typedef __attribute__((ext_vector_type(2))) float v2f;
typedef __attribute__((ext_vector_type(8))) float v8f;
typedef __attribute__((ext_vector_type(4))) unsigned int u32x4;
typedef __attribute__((ext_vector_type(8))) int i32x8;
typedef __attribute__((ext_vector_type(4))) int i32x4;

// D = A(16x4) * B(4x16) + C, fp32 WMMA (wave32). 8-arg VOP3P form.
__device__ __forceinline__ v8f wmma4(v2f a, v2f b, v8f c) {
  return __builtin_amdgcn_wmma_f32_16x16x4_f32(
      false, a, false, b, (short)0, c, false, false);
}

__device__ __forceinline__ float reluf(float x) { return x > 0.f ? x : 0.f; }

__device__ __forceinline__ float wave_sum(float v) {
#pragma unroll
  for (int o = 16; o > 0; o >>= 1) v += __shfl_down(v, o, 32);
  return v;
}

// ---------------------------------------------------------------------------
// TDM: load a 256x256 f32 row-major matrix from global into LDS offset 0.
// D# bit layout per CDNA5 ISA ch.8: group0 {count=1, lds_addr=0, global_addr,
// type=2}; group1 {data_size=4B, tensor_dim0/1=256, tile_dim0/1=256,
// dim0_stride=256}; groups 2/3 benign (dims=1). This toolchain exposes the
// 6-arg builtin (extra int32x8 group slot before cpol) — pass it zeroed.
// Issued by one thread; the issuing wave waits on TENSORcnt, workgroup syncs.
// ---------------------------------------------------------------------------
__device__ __forceinline__ void tdm_load_A_to_lds0(const float* gsrc) {
  const unsigned long long ga = (unsigned long long)gsrc;
  const u32x4 g0 = {
      1u,                                   // count=1, user descriptor
      0u,                                   // lds_addr = 0 (dyn-LDS base)
      (unsigned)(ga & 0xFFFFFFFFu),         // global_addr[31:0]
      (unsigned)((ga >> 32) & 0x01FFFFFFu)  // global_addr[56:32]
          | (2u << 30)};                    // type = 2 ("image")
  const i32x8 g1 = {
      (int)(2u << 16),   // wg_mask=0, data_size=2 (4 bytes)
      (int)(NX << 16),   // tensor_dim0[15:0] in bits[63:48]
      (int)(NX << 16),   // dim0[31:16]=0 | tensor_dim1[15:0]
      (int)(NX << 16),   // dim1[31:16]=0 | tile_dim0 = 256
      (int)NX,           // tile_dim1 = 256, tile_dim2 = 0
      (int)NX,           // tensor_dim0_stride[31:0] = 256
      0,                 // stride0[47:32]=0 | tensor_dim1_stride lo
      0};                // tensor_dim1_stride hi
  const i32x4 g2 = {1, 1, 0, 0};                // tensor_dim2=1, tensor_dim3=1
  const i32x4 g3 = {0, (int)(1u << 16), 0, 0};  // tensor_dim4=1
  const i32x8 gx = {0, 0, 0, 0, 0, 0, 0, 0};    // unused extra group slot
  __builtin_amdgcn_tensor_load_to_lds(g0, g1, g2, g3, gx, 0);
}

// ---------------------------------------------------------------------------
// Kernel 0: zero the global accumulator slots.
// ---------------------------------------------------------------------------
__global__ void k_zero(float* acc) {
  if (threadIdx.x < 8) acc[threadIdx.x] = 0.f;
}

// ---------------------------------------------------------------------------
// Kernel 1: one block = one timestep (64 rows). U = HEAT*M*DT (store),
// drive = U@Bw^T + D@Ew^T (store to ws), accumulate U/dx penalties.
// 16 waves x 4 n-tiles sharing the A-fragment (same state rows).
// ---------------------------------------------------------------------------
__global__ __launch_bounds__(512) void k_drive(
    const float* __restrict__ Mf, const float* __restrict__ DTm,
    const float* __restrict__ Dm, const float* __restrict__ Bw,
    const float* __restrict__ Ew, float* __restrict__ Uout,
    float* __restrict__ drive, float* __restrict__ acc) {
  __shared__ float sU[B_DIM * NU];
  __shared__ float sD[B_DIM * ND];
  const long r0 = (long)blockIdx.x * B_DIM;

  float upen = 0.f;
  for (int i = threadIdx.x; i < B_DIM * NU; i += 512) {
    const long g = r0 * NU + i;
    const float uv = HEAT_CONST * Mf[g] * DTm[g];
    sU[i] = uv;
    Uout[g] = uv;
    upen += reluf(UMIN_C - uv) + reluf(uv - UMAX_C);
    sD[i] = Dm[g];
  }
  __syncthreads();

  const int lane = threadIdx.x & 31;
  const int wave = threadIdx.x >> 5;
  const int m = lane & 15;          // A row / B col / C col
  const int half = lane >> 4;
  const int kb = half * 2;          // K sub-offset per ISA layout
  const int m0 = (wave >> 2) * 16;
  const int nbase = (wave & 3) * 64;

  v8f bu[4] = {{}, {}, {}, {}};
  v8f ed[4] = {{}, {}, {}, {}};
  for (int kk = 0; kk < NU; kk += 4) {
    v2f a;
    a.x = sU[(m0 + m) * NU + kk + kb];
    a.y = sU[(m0 + m) * NU + kk + kb + 1];
#pragma unroll
    for (int q = 0; q < 4; ++q) {
      const int n0 = nbase + q * 16;
      v2f b;
      b.x = Bw[(n0 + m) * NU + kk + kb];      // (Bw^T)[k][n] = Bw[n][k]
      b.y = Bw[(n0 + m) * NU + kk + kb + 1];
      bu[q] = wmma4(a, b, bu[q]);
    }
  }
  for (int kk = 0; kk < ND; kk += 4) {
    v2f a;
    a.x = sD[(m0 + m) * ND + kk + kb];
    a.y = sD[(m0 + m) * ND + kk + kb + 1];
#pragma unroll
    for (int q = 0; q < 4; ++q) {
      const int n0 = nbase + q * 16;
      v2f b;
      b.x = Ew[(n0 + m) * ND + kk + kb];
      b.y = Ew[(n0 + m) * ND + kk + kb + 1];
      ed[q] = wmma4(a, b, ed[q]);
    }
  }

  float dxu = 0.f, dxd = 0.f;
#pragma unroll
  for (int q = 0; q < 4; ++q) {
    const int n0 = nbase + q * 16;
#pragma unroll
    for (int i = 0; i < 8; ++i) {
      const float bv = bu[q][i], ev = ed[q][i];
      dxu += reluf(DXMIN_C - bv) + reluf(bv - DXMAX_C);
      dxd += reluf(DXMIN_C - ev) + reluf(ev - DXMAX_C);
      drive[(r0 + m0 + i + 8 * half) * NX + n0 + m] = bv + ev;
    }
  }
  upen = wave_sum(upen);
  dxu = wave_sum(dxu);
  dxd = wave_sum(dxd);
  if (lane == 0) {
    atomicAdd(&acc[0], upen);
    atomicAdd(&acc[1], dxu);
    atomicAdd(&acc[2], dxd);
  }
}

// ---------------------------------------------------------------------------
// Kernel 2: dst = src @ src (256x256x256) — repeated-squaring step for A^32.
// ---------------------------------------------------------------------------
__global__ __launch_bounds__(256) void k_square(const float* __restrict__ src,
                                                float* __restrict__ dst) {
  const int lane = threadIdx.x & 31;
  const int wave = threadIdx.x >> 5;
  const int tile = blockIdx.x * 8 + wave;
  const int m0 = (tile >> 4) * 16, n0 = (tile & 15) * 16;
  const int m = lane & 15, half = lane >> 4, kb = half * 2;
  v8f c = {};
  for (int kk = 0; kk < NX; kk += 4) {
    v2f a, b;
    a.x = src[(m0 + m) * NX + kk + kb];
    a.y = src[(m0 + m) * NX + kk + kb + 1];
    b.x = src[(kk + kb) * NX + n0 + m];
    b.y = src[(kk + kb + 1) * NX + n0 + m];
    c = wmma4(a, b, c);
  }
#pragma unroll
  for (int i = 0; i < 8; ++i)
    dst[(m0 + i + 8 * half) * NX + n0 + m] = c[i];
}

// ---------------------------------------------------------------------------
// Kernel 3: per-chunk local scan from zero state -> s_j (end-of-chunk state).
// A via TDM into the first 256KB of LDS; state at +256KB. 320KB total, 1 WG per WGP.
// Each wave: 4 n-tiles sharing the A-fragment; drive prefetched one step ahead.
// ---------------------------------------------------------------------------
__global__ __launch_bounds__(512) void k_scan_local(
    const float* __restrict__ Aw, const float* __restrict__ drive,
    float* __restrict__ s_out) {
  extern __shared__ float smem[];
  float* sA = smem;             // NX*NX  (LDS offset 0 -> TDM target)
  float* sS = smem + NX * NX;   // B_DIM*NX
  if (threadIdx.x == 0) tdm_load_A_to_lds0(Aw);
  for (int i = threadIdx.x; i < B_DIM * NX; i += 512) sS[i] = 0.f;
  if (threadIdx.x == 0) __builtin_amdgcn_s_wait_tensorcnt(0);
  __syncthreads();

  const int lane = threadIdx.x & 31;
  const int wave = threadIdx.x >> 5;
  const int m = lane & 15, half = lane >> 4, kb = half * 2;
  const int m0 = (wave >> 2) * 16;
  const int nbase = (wave & 3) * 64;
  const long t0 = (long)blockIdx.x * CHUNK_LEN;

  v8f nxt[4];
#pragma unroll
  for (int q = 0; q < 4; ++q) {
    const float* drv = drive + (t0 * B_DIM + m0) * NX + nbase + q * 16;
#pragma unroll
    for (int i = 0; i < 8; ++i) nxt[q][i] = drv[(i + 8 * half) * NX + m];
  }

  for (int s = 0; s < CHUNK_LEN; ++s) {
    v8f c[4];
#pragma unroll
    for (int q = 0; q < 4; ++q) c[q] = nxt[q];
    if (s + 1 < CHUNK_LEN) {  // prefetch next step's drive during the K-loop
      const long t1 = t0 + s + 1;
#pragma unroll
      for (int q = 0; q < 4; ++q) {
        const float* drv = drive + (t1 * B_DIM + m0) * NX + nbase + q * 16;
#pragma unroll
        for (int i = 0; i < 8; ++i) nxt[q][i] = drv[(i + 8 * half) * NX + m];
      }
    }
    for (int kk = 0; kk < NX; kk += 4) {
      v2f a;
      a.x = sS[(m0 + m) * NX + kk + kb];
      a.y = sS[(m0 + m) * NX + kk + kb + 1];
#pragma unroll
      for (int q = 0; q < 4; ++q) {
        const int n0 = nbase + q * 16;
        v2f b;
        b.x = sA[(n0 + m) * NX + kk + kb];    // (A^T)[k][n] = A[n][k]
        b.y = sA[(n0 + m) * NX + kk + kb + 1];
        c[q] = wmma4(a, b, c[q]);
      }
    }
    __syncthreads();
#pragma unroll
    for (int q = 0; q < 4; ++q) {
      const int n0 = nbase + q * 16;
#pragma unroll
      for (int i = 0; i < 8; ++i)
        sS[(m0 + i + 8 * half) * NX + n0 + m] = c[q][i];
    }
    __syncthreads();
  }
  float* dst = s_out + (long)blockIdx.x * B_DIM * NX;
  for (int i = threadIdx.x; i < B_DIM * NX; i += 512) dst[i] = sS[i];
}

// ---------------------------------------------------------------------------
// Kernel 4: sequential carry propagation: carry_{j+1} = carry_j @ (A^32)^T + s_j.
// Single workgroup; P staged via TDM. Stores carry_j for every chunk.
// ---------------------------------------------------------------------------
__global__ __launch_bounds__(512) void k_carry(
    const float* __restrict__ x_in, const float* __restrict__ x0c,
    const float* __restrict__ P, const float* __restrict__ s_in,
    float* __restrict__ carries) {
  extern __shared__ float smem[];
  float* sP = smem;
  float* sS = smem + NX * NX;
  if (threadIdx.x == 0) tdm_load_A_to_lds0(P);
  for (int i = threadIdx.x; i < B_DIM * NX; i += 512)
    sS[i] = x_in[i] + x0c[i % NX];   // carry_0 = x + x0_correct
  if (threadIdx.x == 0) __builtin_amdgcn_s_wait_tensorcnt(0);
  __syncthreads();

  const int lane = threadIdx.x & 31;
  const int wave = threadIdx.x >> 5;
  const int m = lane & 15, half = lane >> 4, kb = half * 2;
  const int m0 = (wave >> 2) * 16;
  const int nbase = (wave & 3) * 64;

  for (int j = 0; j < NCHUNK; ++j) {
    float* cdst = carries + (long)j * B_DIM * NX;
    for (int i = threadIdx.x; i < B_DIM * NX; i += 512) cdst[i] = sS[i];
    if (j == NCHUNK - 1) break;

    const float* sj = s_in + (long)j * B_DIM * NX;
    v8f c[4];
#pragma unroll
    for (int q = 0; q < 4; ++q) {
      const int n0 = nbase + q * 16;
#pragma unroll
      for (int i = 0; i < 8; ++i)
        c[q][i] = sj[(m0 + i + 8 * half) * NX + n0 + m];
    }
    for (int kk = 0; kk < NX; kk += 4) {
      v2f a;
      a.x = sS[(m0 + m) * NX + kk + kb];
      a.y = sS[(m0 + m) * NX + kk + kb + 1];
#pragma unroll
      for (int q = 0; q < 4; ++q) {
        const int n0 = nbase + q * 16;
        v2f b;
        b.x = sP[(n0 + m) * NX + kk + kb];
        b.y = sP[(n0 + m) * NX + kk + kb + 1];
        c[q] = wmma4(a, b, c[q]);
      }
    }
    __syncthreads();
#pragma unroll
    for (int q = 0; q < 4; ++q) {
      const int n0 = nbase + q * 16;
#pragma unroll
      for (int i = 0; i < 8; ++i)
        sS[(m0 + i + 8 * half) * NX + n0 + m] = c[q][i];
    }
    __syncthreads();
  }
}

// ---------------------------------------------------------------------------
// Kernel 5: per-chunk re-scan from correct carry; emit X, Y; accumulate
// X-bound penalties and sum((X - X_prev)^2).
// ---------------------------------------------------------------------------
__global__ __launch_bounds__(512) void k_scan_emit(
    const float* __restrict__ Aw, const float* __restrict__ drive,
    const float* __restrict__ carries, const float* __restrict__ Cw,
    float* __restrict__ Xout, float* __restrict__ Yout,
    float* __restrict__ acc_g) {
  extern __shared__ float smem[];
  float* sA = smem;
  float* sS = smem + NX * NX;
  if (threadIdx.x == 0) tdm_load_A_to_lds0(Aw);
  {
    const float* csrc = carries + (long)blockIdx.x * B_DIM * NX;
    for (int i = threadIdx.x; i < B_DIM * NX; i += 512) sS[i] = csrc[i];
  }
  if (threadIdx.x == 0) __builtin_amdgcn_s_wait_tensorcnt(0);
  __syncthreads();

  const int lane = threadIdx.x & 31;
  const int wave = threadIdx.x >> 5;
  const int m = lane & 15, half = lane >> 4, kb = half * 2;
  const int m0 = (wave >> 2) * 16;
  const int nbase = (wave & 3) * 64;
  const long t0 = (long)blockIdx.x * CHUNK_LEN;

  float p_sxmin = 0.f, p_sxmax = 0.f, p_sdx = 0.f;

  v8f nxt[4];
#pragma unroll
  for (int q = 0; q < 4; ++q) {
    const float* drv = drive + (t0 * B_DIM + m0) * NX + nbase + q * 16;
#pragma unroll
    for (int i = 0; i < 8; ++i) nxt[q][i] = drv[(i + 8 * half) * NX + m];
  }

  for (int s = 0; s < CHUNK_LEN; ++s) {
    const long t = t0 + s;
    v8f c[4];
#pragma unroll
    for (int q = 0; q < 4; ++q) c[q] = nxt[q];
    if (s + 1 < CHUNK_LEN) {
      const long t1 = t + 1;
#pragma unroll
      for (int q = 0; q < 4; ++q) {
        const float* drv = drive + (t1 * B_DIM + m0) * NX + nbase + q * 16;
#pragma unroll
        for (int i = 0; i < 8; ++i) nxt[q][i] = drv[(i + 8 * half) * NX + m];
      }
    }
    for (int kk = 0; kk < NX; kk += 4) {
      v2f a;
      a.x = sS[(m0 + m) * NX + kk + kb];
      a.y = sS[(m0 + m) * NX + kk + kb + 1];
#pragma unroll
      for (int q = 0; q < 4; ++q) {
        const int n0 = nbase + q * 16;
        v2f b;
        b.x = sA[(n0 + m) * NX + kk + kb];
        b.y = sA[(n0 + m) * NX + kk + kb + 1];
        c[q] = wmma4(a, b, c[q]);
      }
    }
    // penalties + X output + sdx against old state (still in LDS)
#pragma unroll
    for (int q = 0; q < 4; ++q) {
      const int n0 = nbase + q * 16;
      float* xr = Xout + (t * B_DIM + m0) * NX + n0;
#pragma unroll
      for (int i = 0; i < 8; ++i) {
        const int row = i + 8 * half;
        const float nv = c[q][i];
        const float ov = sS[(m0 + row) * NX + n0 + m];  // X_prev
        const float d = nv - ov;
        p_sdx += d * d;
        p_sxmin += reluf(XMIN_C - nv);
        p_sxmax += reluf(nv - XMAX_C);
        xr[row * NX + m] = nv;
      }
    }
    __syncthreads();
#pragma unroll
    for (int q = 0; q < 4; ++q) {
      const int n0 = nbase + q * 16;
#pragma unroll
      for (int i = 0; i < 8; ++i)
        sS[(m0 + i + 8 * half) * NX + n0 + m] = c[q][i];
    }
    __syncthreads();

    // Y_t = X_t @ Cw^T : 4x4 tile grid, one tile per wave.
    {
      const int ym0 = (wave >> 2) * 16, yn0 = (wave & 3) * 16;
      v8f yc = {};
      for (int kk = 0; kk < NX; kk += 4) {
        v2f a, b;
        a.x = sS[(ym0 + m) * NX + kk + kb];
        a.y = sS[(ym0 + m) * NX + kk + kb + 1];
        b.x = Cw[(yn0 + m) * NX + kk + kb];   // (Cw^T)[k][n] = Cw[n][k]
        b.y = Cw[(yn0 + m) * NX + kk + kb + 1];
        yc = wmma4(a, b, yc);
      }
      float* yr = Yout + (t * B_DIM + ym0) * NY + yn0;
#pragma unroll
      for (int i = 0; i < 8; ++i)
        yr[(i + 8 * half) * NY + m] = yc[i];
    }
  }

  p_sxmin = wave_sum(p_sxmin);
  p_sxmax = wave_sum(p_sxmax);
  p_sdx = wave_sum(p_sdx);
  if (lane == 0) {
    atomicAdd(&acc_g[3], p_sxmin);
    atomicAdd(&acc_g[4], p_sxmax);
    atomicAdd(&acc_g[5], p_sdx);
  }
}

// ---------------------------------------------------------------------------
// Kernel 6: combine accumulators into the scalar reg output.
// ---------------------------------------------------------------------------
__global__ void k_finalize(const float* __restrict__ acc,
                           float* __restrict__ reg_out) {
  const float cntX = (float)T_DIM * (float)B_DIM * (float)NX;
  const float cntU = (float)T_DIM * (float)B_DIM * (float)NU;
  const float sumu = acc[0] / cntU;             // sumin + sumax
  const float dxu = acc[1] / cntX;
  const float dxd = acc[2] / cntX;
  const float sx = (acc[3] + acc[4]) / cntX;    // sxmin + sxmax
  const float sdx = acc[5] / cntX;
  reg_out[0] = (Q_CON_X_C / NX) * sx + (Q_CON_U_C / NU) * sumu +
               (Q_DX_C / NX) * sdx + (Q_DX_UD_C / NX) * (dxu + dxd);
}

// ---------------------------------------------------------------------------
extern "C" void kernel_launch(void* const* d_in, const int* in_sizes, int n_in,
                              void* d_out, int out_size, void* d_ws,
                              size_t ws_size, hipStream_t stream) {
  const float* x = (const float*)d_in[0];
  const float* Mf = (const float*)d_in[1];
  const float* DTm = (const float*)d_in[2];
  const float* Dm = (const float*)d_in[3];
  const float* Aw = (const float*)d_in[4];
  const float* Bw = (const float*)d_in[5];
  const float* Ew = (const float*)d_in[6];
  const float* Cw = (const float*)d_in[7];
  const float* x0c = (const float*)d_in[8];

  float* Xout = (float*)d_out;                          // T*B*NX
  float* Yout = Xout + (size_t)T_DIM * B_DIM * NX;      // T*B*NY
  float* Uout = Yout + (size_t)T_DIM * B_DIM * NY;      // T*B*NU
  float* Rout = Uout + (size_t)T_DIM * B_DIM * NU;      // scalar reg

  float* ws = (float*)d_ws;
  float* drive = ws;                                     // T*B*NX
  float* s_buf = drive + (size_t)T_DIM * B_DIM * NX;     // NCHUNK*B*NX
  float* carries = s_buf + (size_t)NCHUNK * B_DIM * NX;  // NCHUNK*B*NX
  float* pow0 = carries + (size_t)NCHUNK * B_DIM * NX;   // NX*NX
  float* pow1 = pow0 + (size_t)NX * NX;                  // NX*NX
  float* acc = pow1 + (size_t)NX * NX;                   // 8 slots

  const int big_lds = (NX * NX + B_DIM * NX) * (int)sizeof(float);  // 320 KB
  (void)hipFuncSetAttribute((const void*)k_scan_local,
                            hipFuncAttributeMaxDynamicSharedMemorySize,
                            big_lds);
  (void)hipFuncSetAttribute((const void*)k_carry,
                            hipFuncAttributeMaxDynamicSharedMemorySize,
                            big_lds);
  (void)hipFuncSetAttribute((const void*)k_scan_emit,
                            hipFuncAttributeMaxDynamicSharedMemorySize,
                            big_lds);

  k_zero<<<1, 32, 0, stream>>>(acc);
  k_drive<<<T_DIM, 512, 0, stream>>>(Mf, DTm, Dm, Bw, Ew, Uout, drive, acc);
  // A^32 via repeated squaring (ping-pong): final power lands in pow0.
  k_square<<<32, 256, 0, stream>>>(Aw, pow0);    // A^2
  k_square<<<32, 256, 0, stream>>>(pow0, pow1);  // A^4
  k_square<<<32, 256, 0, stream>>>(pow1, pow0);  // A^8
  k_square<<<32, 256, 0, stream>>>(pow0, pow1);  // A^16
  k_square<<<32, 256, 0, stream>>>(pow1, pow0);  // A^32

  k_scan_local<<<NCHUNK, 512, big_lds, stream>>>(Aw, drive, s_buf);
  k_carry<<<1, 512, big_lds, stream>>>(x, x0c, pow0, s_buf, carries);
  k_scan_emit<<<NCHUNK, 512, big_lds, stream>>>(Aw, drive, carries, Cw, Xout,
                                                Yout, acc);
  k_finalize<<<1, 1, 0, stream>>>(acc, Rout);
}